// PassingGNN_49555332661734
// MI455X (gfx1250) — compile-verified
//
#include <hip/hip_runtime.h>
#include <hip/hip_bf16.h>

// ---------------------------------------------------------------------------
// 2-layer GAT (heads=1) for MI455X / gfx1250.
//   Dense GEMMs: V_WMMA_F32_16X16X4_F32 (fp32 matrix pipe; GEMM is ~1 GFLOP,
//   negligible, so keep full precision).
//   Sparse edge phase: 3-pass segment softmax + feature-parallel atomic
//   scatter; all accumulators (12.8 MB) are L2-resident (192 MB L2).
// ---------------------------------------------------------------------------

typedef __attribute__((ext_vector_type(2))) float v2f;
typedef __attribute__((ext_vector_type(8))) float v8f;

__device__ __forceinline__ v8f wmma_f32x4(v2f a, v2f b, v8f c) {
#if defined(__HIP_DEVICE_COMPILE__)
#if __has_builtin(__builtin_amdgcn_wmma_f32_16x16x4_f32)
  // 8 args: (neg_a, A, neg_b, B, c_mod, C, reuse_a, reuse_b)
  return __builtin_amdgcn_wmma_f32_16x16x4_f32(
      false, a, false, b, (short)0, c, false, false);
#else
  asm volatile("v_wmma_f32_16x16x4_f32 %0, %1, %2, %0"
               : "+v"(c) : "v"(a), "v"(b));
  return c;
#endif
#else
  (void)a; (void)b;
  return c;
#endif
}

// ---------------------------------------------------------------------------
// GEMM: H[N, NOUT] = X[N, K] @ W[K, NOUT], via V_WMMA_F32_16X16X4_F32.
// One wave per 16-row tile; W staged in LDS.
// A-frag (16x4 f32): lanes 0-15 hold K=k,k+1; lanes 16-31 hold K=k+2,k+3.
// B-frag (4x16 f32): VGPR0 = rows {k, k+2}, VGPR1 = rows {k+1, k+3},
//                    N = lane%16 (mirrors C layout: VGPR i <-> rows {i, i+R/2}).
// C (16x16 f32): VGPR i holds M = i + 8*(lane/16), N = lane%16.
// ---------------------------------------------------------------------------
template <int K, int NOUT>
__global__ __launch_bounds__(256) void gat_gemm_wmma(
    const float* __restrict__ X, const float* __restrict__ W,
    float* __restrict__ H, int nTiles) {
  __shared__ float ldsW[K * NOUT];
  for (int i = threadIdx.x; i < K * NOUT; i += blockDim.x) ldsW[i] = W[i];
  __syncthreads();

  const int lane       = threadIdx.x & 31;
  const int waveInBlk  = threadIdx.x >> 5;
  const int wavesPerBlk = blockDim.x >> 5;   // 8
  const int half = lane >> 4;                // 0 or 1
  const int mrow = lane & 15;
  constexpr int NT = NOUT / 16;

  for (int tile = blockIdx.x * wavesPerBlk + waveInBlk; tile < nTiles;
       tile += gridDim.x * wavesPerBlk) {
    const int rowBase = tile * 16;
    const float* __restrict__ xrow = X + (size_t)(rowBase + mrow) * K;

    v8f c[NT];
#pragma unroll
    for (int t = 0; t < NT; ++t) {
#pragma unroll
      for (int q = 0; q < 8; ++q) c[t][q] = 0.0f;
    }

    for (int k = 0; k < K; k += 4) {
      v2f a;
      a.x = xrow[k + 2 * half];
      a.y = xrow[k + 2 * half + 1];
#pragma unroll
      for (int t = 0; t < NT; ++t) {
        v2f b;
        b.x = ldsW[(k + 2 * half) * NOUT + t * 16 + mrow];
        b.y = ldsW[(k + 2 * half + 1) * NOUT + t * 16 + mrow];
        c[t] = wmma_f32x4(a, b, c[t]);
      }
    }

#pragma unroll
    for (int t = 0; t < NT; ++t) {
#pragma unroll
      for (int i = 0; i < 8; ++i) {
        const int r = rowBase + i + 8 * half;
        H[(size_t)r * NOUT + t * 16 + mrow] = c[t][i];
      }
    }
  }
}

// Scalar tail for N % 16 != 0 (not used for N=100000, kept for generality).
template <int K, int NOUT>
__global__ void gat_gemm_tail(const float* __restrict__ X,
                              const float* __restrict__ W,
                              float* __restrict__ H, int rowStart, int N) {
  const int idx = blockIdx.x * blockDim.x + threadIdx.x;
  const int total = (N - rowStart) * NOUT;
  if (idx >= total) return;
  const int r = rowStart + idx / NOUT;
  const int col = idx % NOUT;
  float acc = 0.0f;
  for (int k = 0; k < K; ++k) acc += X[(size_t)r * K + k] * W[k * NOUT + col];
  H[(size_t)r * NOUT + col] = acc;
}

// Per-node attention scores: s_src[i] = h_i . a_src ; s_dst[i] = h_i . a_dst
template <int F>
__global__ __launch_bounds__(256) void gat_score(
    const float* __restrict__ H, const float* __restrict__ asrc,
    const float* __restrict__ adst, float* __restrict__ ssrc,
    float* __restrict__ sdst, int N) {
  const int i = blockIdx.x * blockDim.x + threadIdx.x;
  if (i >= N) return;
  const float* __restrict__ h = H + (size_t)i * F;
  float s1 = 0.0f, s2 = 0.0f;
#pragma unroll
  for (int f = 0; f < F; ++f) {
    const float v = h[f];
    s1 += v * asrc[f];
    s2 += v * adst[f];
  }
  ssrc[i] = s1;
  sdst[i] = s2;
}

__global__ __launch_bounds__(256) void gat_init_mz(float* __restrict__ m,
                                                   float* __restrict__ z,
                                                   int N) {
  const int i = blockIdx.x * blockDim.x + threadIdx.x;
  if (i < N) {
    m[i] = -__builtin_inff();
    z[i] = 0.0f;
  }
}

template <int F>
__global__ __launch_bounds__(256) void gat_init_bias(
    float* __restrict__ out, const float* __restrict__ bias, int N) {
  const int i = blockIdx.x * blockDim.x + threadIdx.x;
  if (i < N * F) out[i] = bias[i & (F - 1)];
}

// f32 atomic max via signed/unsigned int atomics (works with -inf init).
__device__ __forceinline__ void atomicMaxFloat(float* addr, float v) {
  if (v >= 0.0f)
    atomicMax((int*)addr, __float_as_int(v));
  else
    atomicMin((unsigned int*)addr, __float_as_uint(v));
}

__device__ __forceinline__ void edge_pair(const int* __restrict__ ei,
                                          long long E, long long i, int& s,
                                          int& d) {
  if (i < E) {
    s = ei[i];
    d = ei[E + i];
  } else {  // self loop appended after real edges
    s = d = (int)(i - E);
  }
}

// Pass 1: e = LeakyReLU(s_src[src] + s_dst[dst]); store e; m[dst] = max(...)
__global__ __launch_bounds__(256) void gat_edge_max(
    const int* __restrict__ ei, long long E, long long ET,
    const float* __restrict__ ssrc, const float* __restrict__ sdst,
    float* __restrict__ ebuf, float* __restrict__ m) {
  for (long long i = blockIdx.x * (long long)blockDim.x + threadIdx.x; i < ET;
       i += (long long)gridDim.x * blockDim.x) {
    int s, d;
    edge_pair(ei, E, i, s, d);
    float e = ssrc[s] + sdst[d];
    e = (e > 0.0f) ? e : 0.2f * e;  // NEG_SLOPE = 0.2
    ebuf[i] = e;
    atomicMaxFloat(&m[d], e);
  }
}

// Pass 2: ex = exp(e - m[dst]) (in place); z[dst] += ex
__global__ __launch_bounds__(256) void gat_edge_expsum(
    const int* __restrict__ ei, long long E, long long ET,
    float* __restrict__ ebuf, const float* __restrict__ m,
    float* __restrict__ z) {
  for (long long i = blockIdx.x * (long long)blockDim.x + threadIdx.x; i < ET;
       i += (long long)gridDim.x * blockDim.x) {
    int s, d;
    edge_pair(ei, E, i, s, d);
    const float ex = expf(ebuf[i] - m[d]);
    ebuf[i] = ex;
    atomicAdd(&z[d], ex);
  }
}

// Pass 3: out[dst, f] += (ex / z[dst]) * h[src, f], one lane per feature.
template <int F>
__global__ __launch_bounds__(256) void gat_edge_scatter(
    const int* __restrict__ ei, long long E, long long ET,
    const float* __restrict__ ebuf, const float* __restrict__ z,
    const float* __restrict__ H, float* __restrict__ out) {
  constexpr int LF = (F == 32) ? 5 : 4;
  const long long total = ET << LF;
  for (long long t = blockIdx.x * (long long)blockDim.x + threadIdx.x;
       t < total; t += (long long)gridDim.x * blockDim.x) {
    const long long i = t >> LF;
    const int f = (int)(t & (F - 1));
    int s, d;
    edge_pair(ei, E, i, s, d);
    const float w = ebuf[i] / z[d];
    atomicAdd(out + (size_t)d * F + f, w * H[(size_t)s * F + f]);
  }
}

__global__ __launch_bounds__(256) void gat_elu(float* __restrict__ a,
                                               long long n) {
  for (long long i = blockIdx.x * (long long)blockDim.x + threadIdx.x; i < n;
       i += (long long)gridDim.x * blockDim.x) {
    const float v = a[i];
    a[i] = (v > 0.0f) ? v : (expf(v) - 1.0f);
  }
}

// ---------------------------------------------------------------------------

extern "C" void kernel_launch(void* const* d_in, const int* in_sizes, int n_in,
                              void* d_out, int out_size, void* d_ws,
                              size_t ws_size, hipStream_t stream) {
  const float* x     = (const float*)d_in[0];
  const int*   ei    = (const int*)d_in[1];  // edge_index flattened [2, E]
  const float* W1    = (const float*)d_in[2];
  const float* asrc1 = (const float*)d_in[3];
  const float* adst1 = (const float*)d_in[4];
  const float* b1    = (const float*)d_in[5];
  const float* W2    = (const float*)d_in[6];
  const float* asrc2 = (const float*)d_in[7];
  const float* adst2 = (const float*)d_in[8];
  const float* b2    = (const float*)d_in[9];
  float* out = (float*)d_out;

  const int N = in_sizes[0] / 128;
  const long long E = in_sizes[1] / 2;
  const long long ET = E + N;  // with self loops

  // Workspace layout (floats). h1 region reused for layer-2 features (N*16).
  float* p    = (float*)d_ws;
  float* h1   = p; p += (size_t)N * 32;
  float* acc  = p; p += (size_t)N * 32;  // layer-1 aggregation, ELU'd in place
  float* ssrc = p; p += N;
  float* sdst = p; p += N;
  float* m    = p; p += N;
  float* z    = p; p += N;
  float* ebuf = p; p += ET;
  (void)ws_size;

  const int nTiles = N / 16;
  const int tailRows = N - nTiles * 16;
  const int gemmBlocks = (nTiles + 7) / 8;      // 8 waves / block
  const int nodeBlocks = (N + 255) / 256;
  const int edgeBlocks = (int)((ET + 255) / 256);

  // ---------------- Layer 1: 128 -> 32 ----------------
  gat_gemm_wmma<128, 32><<<gemmBlocks, 256, 0, stream>>>(x, W1, h1, nTiles);
  if (tailRows > 0)
    gat_gemm_tail<128, 32><<<(tailRows * 32 + 255) / 256, 256, 0, stream>>>(
        x, W1, h1, nTiles * 16, N);
  gat_score<32><<<nodeBlocks, 256, 0, stream>>>(h1, asrc1, adst1, ssrc, sdst, N);
  gat_init_mz<<<nodeBlocks, 256, 0, stream>>>(m, z, N);
  gat_init_bias<32><<<(N * 32 + 255) / 256, 256, 0, stream>>>(acc, b1, N);
  gat_edge_max<<<edgeBlocks, 256, 0, stream>>>(ei, E, ET, ssrc, sdst, ebuf, m);
  gat_edge_expsum<<<edgeBlocks, 256, 0, stream>>>(ei, E, ET, ebuf, m, z);
  gat_edge_scatter<32><<<(int)((ET * 32 + 255) / 256), 256, 0, stream>>>(
      ei, E, ET, ebuf, z, h1, acc);
  gat_elu<<<(int)(((long long)N * 32 + 255) / 256), 256, 0, stream>>>(
      acc, (long long)N * 32);

  // ---------------- Layer 2: 32 -> 16 ----------------
  float* h2 = h1;  // reuse
  gat_gemm_wmma<32, 16><<<gemmBlocks, 256, 0, stream>>>(acc, W2, h2, nTiles);
  if (tailRows > 0)
    gat_gemm_tail<32, 16><<<(tailRows * 16 + 255) / 256, 256, 0, stream>>>(
        acc, W2, h2, nTiles * 16, N);
  gat_score<16><<<nodeBlocks, 256, 0, stream>>>(h2, asrc2, adst2, ssrc, sdst, N);
  gat_init_mz<<<nodeBlocks, 256, 0, stream>>>(m, z, N);
  gat_init_bias<16><<<(N * 16 + 255) / 256, 256, 0, stream>>>(out, b2, N);
  gat_edge_max<<<edgeBlocks, 256, 0, stream>>>(ei, E, ET, ssrc, sdst, ebuf, m);
  gat_edge_expsum<<<edgeBlocks, 256, 0, stream>>>(ei, E, ET, ebuf, m, z);
  gat_edge_scatter<16><<<(int)((ET * 16 + 255) / 256), 256, 0, stream>>>(
      ei, E, ET, ebuf, z, h2, out);
}